// MPNEncoder_81707457839127
// MI455X (gfx1250) — compile-verified
//
#include <hip/hip_runtime.h>
#include <math.h>

typedef __attribute__((ext_vector_type(2))) float v2f;
typedef __attribute__((ext_vector_type(8))) float v8f;
typedef __attribute__((ext_vector_type(4))) unsigned int v4u;
typedef __attribute__((ext_vector_type(8))) unsigned int v8u;
typedef unsigned long long u64;

// ---------------- problem constants ----------------
static constexpr int H   = 300;
static constexpr int H4  = H / 4;            // 75 float4 per row
static constexpr int B   = 1024;
static constexpr int A   = 50;
static constexpr int NA  = 1 + B * A;        // 51201 atoms
static constexpr int NB  = 1 + B * 2 * A;    // 102401 directed bonds
static constexpr int AF  = 133;
static constexpr int BF  = 147;
static constexpr int MAXNB = 6;

// ---------------- TDM 2-D tile load: global -> LDS, zero-fill OOB ----------------
// Builds D# groups per CDNA5 ISA and issues
// tensor_load_to_lds (2-D: groups 2/3 disabled). Hardware applies LDS padding
// (pad_amount DWORDs after every pad_interval DWORDs) == our bank-swizzle strides.
static __device__ __forceinline__
void tdm_load_2d(unsigned lds_off,          // LDS byte offset of tile start
                 const void* gaddr,         // global address of tile start
                 unsigned tile_d0,          // tile row length (elements)
                 unsigned tile_d1,          // tile rows
                 unsigned tens_d0,          // valid elements per row (OOB -> 0)
                 unsigned tens_d1,          // valid rows (OOB -> 0)
                 u64 stride0,               // row stride in elements
                 unsigned pad_int_code,     // 4: every 32 dwords ; 5: every 64
                 unsigned pad_amt_code)     // 3: 4 dwords ; 15: 16 dwords
{
  const u64 ga = (u64)gaddr;
  v4u g0;
  g0[0] = 1u;                                        // count=1, user mode
  g0[1] = lds_off;                                   // lds_addr
  g0[2] = (unsigned)(ga & 0xffffffffu);              // global_addr[31:0]
  g0[3] = (unsigned)((ga >> 32) & 0x1ffffffu) | (2u << 30);  // addr[56:32]|type=2
  v8u g1;
  g1[0] = (2u << 16)                                 // data_size = 4 bytes
        | (1u << 20)                                 // pad_enable
        | (pad_int_code << 22) | (pad_amt_code << 25);
  g1[1] = (tens_d0 & 0xffffu) << 16;                 // tensor_dim0[15:0]
  g1[2] = (tens_d0 >> 16) | ((tens_d1 & 0xffffu) << 16);
  g1[3] = (tens_d1 >> 16) | (tile_d0 << 16);         // tensor_dim1[31:16]|tile_dim0
  g1[4] = tile_d1 & 0xffffu;                         // tile_dim1 | tile_dim2=0
  g1[5] = (unsigned)(stride0 & 0xffffffffu);         // tensor_dim0_stride[31:0]
  g1[6] = (unsigned)((stride0 >> 32) & 0xffffu);     // stride0[47:32] | stride1=0
  g1[7] = 0u;
  asm volatile("tensor_load_to_lds %0, %1, null, null"
               :: "s"(g0), "s"(g1)
               : "memory");
}

// ---------------- LDS-tiled, register-blocked WMMA fp32 GEMM body ----------------
// O[m][n] = act( sum_k X[m][k] * Wt[k][n] + bias[n] + add[m][n] )
// Wave grid WGM x WGN (8 waves, 256 thr); each wave computes RM x RN 16x16 tiles.
// Block tile: BM = WGM*RM*16, BN = WGN*RN*16 (BN == 64 here).
// Tiles staged by the Tensor Data Mover (one wave issues; TENSORcnt + barrier).
// A frag: lane l, vgpr v -> (M = l%16, K = 2*(l/16)+v)
// B frag: lane l, vgpr v -> (N = l%16, K = 2*(l/16)+v)
// C/D   : lane l, vgpr v -> (M = v + 8*(l/16), N = l%16)
static constexpr int KC   = 32;
static constexpr int XS_S = 36;  // K-stride: 36*l%64 distinct; +2 offset disjoint mod-4
static constexpr int WS_S = 80;  // rows k,k+2 land 32 banks apart -> disjoint halves

template<int WGM, int WGN, int RM, int RN>
__device__ __forceinline__
void gemm_body(int bid, const float* __restrict__ X, int ldx,
               const float* __restrict__ Wt,
               const float* __restrict__ add,
               const float* __restrict__ bias,
               float* __restrict__ O,
               int M, int N, int K, int relu)
{
  constexpr int BM = WGM * RM * 16;
  constexpr int BN = WGN * RN * 16;   // == 64
  __shared__ alignas(16) float Xs[BM * XS_S];
  __shared__ alignas(16) float Ws[KC * WS_S];

  const int ntn = (N + BN - 1) / BN;
  const int m0  = (bid / ntn) * BM;
  const int n0  = (bid % ntn) * BN;
  const int tid  = threadIdx.x;
  const int lane = tid & 31;
  const int wv   = tid >> 5;
  const int wm   = wv / WGN;
  const int wn   = wv % WGN;
  const int l16  = lane & 15;
  const int hi   = lane >> 4;

  const unsigned xs_off = (unsigned)(u64)(const void*)Xs;
  const unsigned ws_off = (unsigned)(u64)(const void*)Ws;

  v8f c[RM][RN];
#pragma unroll
  for (int im = 0; im < RM; ++im)
#pragma unroll
    for (int in = 0; in < RN; ++in) c[im][in] = {};

  for (int k0 = 0; k0 < K; k0 += KC) {
    if (wv == 0) {
      // X tile [BM x KC]: rows padded 32+4 dwords -> XS_S=36
      tdm_load_2d(xs_off, X + (long)m0 * ldx + k0,
                  KC, BM, (unsigned)(K - k0), (unsigned)(M - m0),
                  (u64)ldx, 4u, 3u);
      // W tile [KC x BN]: rows padded 64+16 dwords -> WS_S=80
      tdm_load_2d(ws_off, Wt + (long)k0 * N + n0,
                  BN, KC, (unsigned)(N - n0), (unsigned)(K - k0),
                  (u64)N, 5u, 15u);
      __builtin_amdgcn_s_wait_tensorcnt(0);
    }
    __syncthreads();

    // ---- branch-free inner loop ----
#pragma unroll
    for (int kk = 0; kk < KC; kk += 4) {
      const int ka = kk + 2 * hi;
      v2f a[RM], b[RN];
#pragma unroll
      for (int im = 0; im < RM; ++im) {
        const int r = (wm * RM + im) * 16 + l16;
        a[im].x = Xs[r * XS_S + ka];
        a[im].y = Xs[r * XS_S + ka + 1];
      }
#pragma unroll
      for (int in = 0; in < RN; ++in) {
        const int cn = (wn * RN + in) * 16 + l16;
        b[in].x = Ws[ka * WS_S + cn];
        b[in].y = Ws[(ka + 1) * WS_S + cn];
      }
#pragma unroll
      for (int im = 0; im < RM; ++im)
#pragma unroll
        for (int in = 0; in < RN; ++in)
          c[im][in] = __builtin_amdgcn_wmma_f32_16x16x4_f32(
              false, a[im], false, b[in], (short)0, c[im][in], false, false);
    }
    __syncthreads();
  }

  // ---- epilogue ----
#pragma unroll
  for (int in = 0; in < RN; ++in) {
    const int col = n0 + (wn * RN + in) * 16 + l16;
    const float bv = (bias && col < N) ? bias[col] : 0.f;
#pragma unroll
    for (int im = 0; im < RM; ++im) {
#pragma unroll
      for (int v = 0; v < 8; ++v) {
        const int m = m0 + (wm * RM + im) * 16 + (hi << 3) + v;
        if (m < M && col < N) {
          float val = c[im][in][v] + bv;
          if (add)  val += add[(long)m * N + col];
          if (relu) val = fmaxf(val, 0.f);
          O[(long)m * N + col] = val;
        }
      }
    }
  }
}

template<int WGM, int WGN, int RM, int RN>
__global__ __launch_bounds__(32 * WGM * WGN)
void k_gemm_wmma(const float* __restrict__ X, int ldx,
                 const float* __restrict__ Wt,
                 const float* __restrict__ add,
                 const float* __restrict__ bias,
                 float* __restrict__ O,
                 int M, int N, int K, int relu)
{
  gemm_body<WGM, WGN, RM, RN>(blockIdx.x, X, ldx, Wt, add, bias, O, M, N, K, relu);
}

// ---- batched GRU-step GEMM: blockIdx.y selects one of 4 independent GEMMs ----
__global__ __launch_bounds__(256)
void k_gemm_gru4(const float* __restrict__ x0, const float* __restrict__ x1,
                 const float* __restrict__ x2, const float* __restrict__ x3,
                 int ldx01, int ldx23,
                 const float* __restrict__ wt0, const float* __restrict__ wt1,
                 const float* __restrict__ wt2, const float* __restrict__ wt3,
                 const float* __restrict__ b0, const float* __restrict__ b1,
                 const float* __restrict__ b2, const float* __restrict__ b3,
                 float* __restrict__ o0, float* __restrict__ o1,
                 float* __restrict__ o2, float* __restrict__ o3,
                 int M, int N, int K)
{
  const float* X; const float* Wt; const float* bias; float* O; int ldx;
  switch (blockIdx.y) {          // uniform per block
    case 0:  X = x0; Wt = wt0; bias = b0; O = o0; ldx = ldx01; break;
    case 1:  X = x1; Wt = wt1; bias = b1; O = o1; ldx = ldx01; break;
    case 2:  X = x2; Wt = wt2; bias = b2; O = o2; ldx = ldx23; break;
    default: X = x3; Wt = wt3; bias = b3; O = o3; ldx = ldx23; break;
  }
  gemm_body<4, 2, 1, 2>(blockIdx.x, X, ldx, Wt, nullptr, bias, O, M, N, K, 0);
}

// ---------------- weight transpose: Wt[k][n] = W[n][k] (one-time) ----------------
__global__ void k_transpose(const float* __restrict__ W, float* __restrict__ Wt,
                            int N, int K)
{
  const long idx = (long)blockIdx.x * blockDim.x + threadIdx.x;
  if (idx >= (long)N * K) return;
  const int n = (int)(idx / K), k = (int)(idx % K);
  Wt[(long)k * N + n] = W[idx];
}

// ---------------- neighbor aggregate (float4): sum * max over MAX_NB ----------------
__global__ void k_aggregate(const float* __restrict__ mbond,
                            const int* __restrict__ a2b,
                            float* __restrict__ agg,      // optional output
                            float* __restrict__ matom,    // optional +=
                            int do_add)
{
  const long idx = (long)blockIdx.x * blockDim.x + threadIdx.x;
  if (idx >= (long)NA * H4) return;
  const int i = (int)(idx / H4), q = (int)(idx % H4);
  const float4* mb4 = (const float4*)mbond;
  float sx = 0.f, sy = 0.f, sz = 0.f, sw = 0.f;
  float mxx = -3.4e38f, mxy = -3.4e38f, mxz = -3.4e38f, mxw = -3.4e38f;
#pragma unroll
  for (int nb = 0; nb < MAXNB; ++nb) {
    const int b = a2b[(long)i * MAXNB + nb];
    float4 v = make_float4(0.f, 0.f, 0.f, 0.f);
    if (b != 0) v = mb4[(long)b * H4 + q];   // masked entries contribute 0.0
    sx += v.x; sy += v.y; sz += v.z; sw += v.w;
    mxx = fmaxf(mxx, v.x); mxy = fmaxf(mxy, v.y);
    mxz = fmaxf(mxz, v.z); mxw = fmaxf(mxw, v.w);
  }
  const float4 av = make_float4(sx * mxx, sy * mxy, sz * mxz, sw * mxw);
  if (agg) ((float4*)agg)[idx] = av;
  if (do_add) {
    float4 t = ((float4*)matom)[idx];
    t.x += av.x; t.y += av.y; t.z += av.z; t.w += av.w;
    ((float4*)matom)[idx] = t;
  }
}

// ---------------- bond-update gather (float4) ----------------
__global__ void k_bond_pre(const float* __restrict__ matom,
                           const float* __restrict__ mbond,
                           const int* __restrict__ b2a,
                           const int* __restrict__ b2revb,
                           float* __restrict__ o)
{
  const long idx = (long)blockIdx.x * blockDim.x + threadIdx.x;
  if (idx >= (long)NB * H4) return;
  const int b = (int)(idx / H4), q = (int)(idx % H4);
  const float4 va = ((const float4*)matom)[(long)b2a[b] * H4 + q];
  const float4 vb = ((const float4*)mbond)[(long)b2revb[b] * H4 + q];
  ((float4*)o)[idx] = make_float4(va.x - vb.x, va.y - vb.y, va.z - vb.z, va.w - vb.w);
}

// ---------------- concat [agg | matom | iatom] -> [NA, 3H] (float4) ----------------
__global__ void k_concat3(const float* __restrict__ a, const float* __restrict__ bb,
                          const float* __restrict__ cc, float* __restrict__ o)
{
  const long idx = (long)blockIdx.x * blockDim.x + threadIdx.x;
  if (idx >= (long)NA * 3 * H4) return;
  const int i = (int)(idx / (3 * H4)), j = (int)(idx % (3 * H4));
  const float* src = (j < H4) ? a : ((j < 2 * H4) ? bb : cc);
  const int jj = (j < H4) ? j : ((j < 2 * H4) ? j - H4 : j - 2 * H4);
  ((float4*)o)[idx] = ((const float4*)src)[(long)i * H4 + jj];
}

// ---------------- message = relu(node + gru_bias) (float4) ----------------
__global__ void k_bias_relu(const float* __restrict__ node,
                            const float* __restrict__ bias,
                            float* __restrict__ msg)
{
  const long idx = (long)blockIdx.x * blockDim.x + threadIdx.x;
  if (idx >= (long)NA * H4) return;
  const float4 n4 = ((const float4*)node)[idx];
  const float4 b4 = ((const float4*)bias)[idx % H4];
  ((float4*)msg)[idx] = make_float4(fmaxf(n4.x + b4.x, 0.f), fmaxf(n4.y + b4.y, 0.f),
                                    fmaxf(n4.z + b4.z, 0.f), fmaxf(n4.w + b4.w, 0.f));
}

// ---------------- h0 = per-molecule max over atoms of node (pre-relu) ----------------
__global__ void k_h0(const float* __restrict__ node, float* __restrict__ hf,
                     float* __restrict__ hb)
{
  const long idx = (long)blockIdx.x * blockDim.x + threadIdx.x;
  if (idx >= (long)B * H4) return;
  const int b = (int)(idx / H4), q = (int)(idx % H4);
  float4 m = make_float4(-3.4e38f, -3.4e38f, -3.4e38f, -3.4e38f);
  for (int t = 0; t < A; ++t) {
    const float4 v = ((const float4*)node)[(long)(1 + b * A + t) * H4 + q];
    m.x = fmaxf(m.x, v.x); m.y = fmaxf(m.y, v.y);
    m.z = fmaxf(m.z, v.z); m.w = fmaxf(m.w, v.w);
  }
  ((float4*)hf)[idx] = m;
  ((float4*)hb)[idx] = m;
}

// ---------------- fused GRU gates for both directions ----------------
static __device__ __forceinline__ float sigm(float x) { return 1.f / (1.f + __expf(-x)); }

__global__ void k_gru_gate(const float* __restrict__ xgf, const float* __restrict__ ghf,
                           const float* __restrict__ xgb, const float* __restrict__ ghb,
                           float* __restrict__ hf, float* __restrict__ hb,
                           float* __restrict__ outf, float* __restrict__ outb,
                           int tf, int tb)
{
  const long idx = (long)blockIdx.x * blockDim.x + threadIdx.x;
  if (idx >= 2L * B * H) return;
  const int  dir = (int)(idx / ((long)B * H));
  const long r   = idx % ((long)B * H);
  const int  b   = (int)(r / H), j = (int)(r % H);
  const float* xg = dir ? xgb : xgf;
  const float* gh = dir ? ghb : ghf;
  float* h  = dir ? hb : hf;
  float* ot = dir ? outb : outf;
  const int t = dir ? tb : tf;
  const long base = (long)b * (3 * H);
  const float rg = sigm(xg[base + j]          + gh[base + j]);
  const float zg = sigm(xg[base + H + j]      + gh[base + H + j]);
  const float ng = tanhf(xg[base + 2 * H + j] + rg * gh[base + 2 * H + j]);
  const float hv = h[r];
  const float hn = (1.f - zg) * ng + zg * hv;
  h[r] = hn;
  ot[((long)b * A + t) * H + j] = hn;
}

// ---------------- [NA, 2H] GRU-output concat (row 0 = [msg0|msg0]) (float4) ----------------
__global__ void k_gcat(const float* __restrict__ msg, const float* __restrict__ outf,
                       const float* __restrict__ outb, float* __restrict__ o)
{
  const long idx = (long)blockIdx.x * blockDim.x + threadIdx.x;
  if (idx >= (long)NA * 2 * H4) return;
  const int i = (int)(idx / (2 * H4)), j = (int)(idx % (2 * H4));
  float4 v;
  if (i == 0) {
    v = ((const float4*)msg)[j % H4];
  } else {
    const long r = i - 1;
    v = (j < H4) ? ((const float4*)outf)[r * H4 + j]
                 : ((const float4*)outb)[r * H4 + (j - H4)];
  }
  ((float4*)o)[idx] = v;
}

// ---------------- readout: per-molecule mean (float4) ----------------
__global__ void k_mean(const float* __restrict__ ah, float* __restrict__ out)
{
  const long idx = (long)blockIdx.x * blockDim.x + threadIdx.x;
  if (idx >= (long)B * H4) return;
  const int b = (int)(idx / H4), q = (int)(idx % H4);
  float4 s = make_float4(0.f, 0.f, 0.f, 0.f);
  for (int t = 0; t < A; ++t) {
    const float4 v = ((const float4*)ah)[(long)(1 + b * A + t) * H4 + q];
    s.x += v.x; s.y += v.y; s.z += v.z; s.w += v.w;
  }
  const float inv = 1.f / (float)A;
  ((float4*)out)[idx] = make_float4(s.x * inv, s.y * inv, s.z * inv, s.w * inv);
}

// ---------------- host-side launchers ----------------
static inline unsigned cdiv256(long n) { return (unsigned)((n + 255) / 256); }

static void launch_gemm(const float* X, int ldx, const float* Wt, const float* add,
                        const float* bias, float* O, int M, int N, int K, int relu,
                        hipStream_t stream)
{
  if (M >= 4096) {
    // big GEMMs: 128x64 block tile, 2x2 register blocking per wave
    const unsigned blocks = (unsigned)(((M + 127) / 128) * ((N + 63) / 64));
    k_gemm_wmma<4, 2, 2, 2><<<blocks, 256, 0, stream>>>(X, ldx, Wt, add, bias, O,
                                                        M, N, K, relu);
  } else {
    // small GEMMs: 64x64 block tile, 1x2 register blocking per wave
    const unsigned blocks = (unsigned)(((M + 63) / 64) * ((N + 63) / 64));
    k_gemm_wmma<4, 2, 1, 2><<<blocks, 256, 0, stream>>>(X, ldx, Wt, add, bias, O,
                                                        M, N, K, relu);
  }
}

extern "C" void kernel_launch(void* const* d_in, const int* in_sizes, int n_in,
                              void* d_out, int out_size, void* d_ws, size_t ws_size,
                              hipStream_t stream)
{
  (void)in_sizes; (void)n_in; (void)out_size; (void)ws_size;

  const float* f_atoms  = (const float*)d_in[0];
  const float* f_bonds  = (const float*)d_in[1];
  const int*   a2b      = (const int*)d_in[2];
  const int*   b2a      = (const int*)d_in[3];
  const int*   b2revb   = (const int*)d_in[4];
  const float* W_i_atom = (const float*)d_in[5];
  const float* W_i_bond = (const float*)d_in[6];
  const float* W_h      = (const float*)d_in[7];   // [2, H, H]
  const float* lr_W     = (const float*)d_in[8];   // [H, 3H]
  const float* W_o_W    = (const float*)d_in[9];   // [H, 2H]
  const float* W_o_b    = (const float*)d_in[10];  // [H]
  const float* gru_bias = (const float*)d_in[11];  // [H]
  const float* gru_w_ih = (const float*)d_in[12];  // [2, 3H, H]
  const float* gru_w_hh = (const float*)d_in[13];  // [2, 3H, H]
  const float* gru_b_ih = (const float*)d_in[14];  // [2, 3H]
  const float* gru_b_hh = (const float*)d_in[15];  // [2, 3H]
  float* out = (float*)d_out;                      // [B, H]

  // ---- workspace layout (floats), manual aliasing ----
  float* P = (float*)d_ws;
  size_t off = 0;
  auto take = [&](size_t n) { float* p = P + off; off += n; return p; };

  float* Wt_ia  = take((size_t)AF * H);
  float* Wt_ib  = take((size_t)BF * H);
  float* Wt_h0  = take((size_t)H * H);
  float* Wt_h1  = take((size_t)H * H);
  float* Wt_lr  = take((size_t)3 * H * H);
  float* Wt_o   = take((size_t)2 * H * H);
  float* Wt_ihf = take((size_t)H * 3 * H);
  float* Wt_ihb = take((size_t)H * 3 * H);
  float* Wt_hhf = take((size_t)H * 3 * H);
  float* Wt_hhb = take((size_t)H * 3 * H);
  float* xg_f   = take((size_t)B * 3 * H);
  float* xg_b   = take((size_t)B * 3 * H);
  float* gh_f   = take((size_t)B * 3 * H);
  float* gh_b   = take((size_t)B * 3 * H);
  float* h_f    = take((size_t)B * H);
  float* h_b    = take((size_t)B * H);

  float* R1  = take((size_t)NA * H);           // input_atom  -> later: message
  float* R2  = take((size_t)NA * H);           // message_atom-> later: out_f
  float* R3  = take((size_t)NA * H);           // agg         -> later: out_b
  float* R45 = take((size_t)2 * NB * H);       // input_bond|mbtmp -> C900 -> gcat+ah
  float* R6  = take((size_t)NB * H);           // message_bond -> later: node

  float* iatom = R1;
  float* matom = R2;
  float* agg   = R3;
  float* ibond = R45;
  float* mbtmp = R45 + (size_t)NB * H;
  float* mbond = R6;
  float* C900  = R45;
  float* node  = R6;
  float* msg   = R1;
  float* outf  = R2;
  float* outb  = R3;
  float* gcat  = R45;
  float* ah    = R45 + (size_t)NA * 2 * H;

  // ---- 1) transpose all weights to k-major ----
  k_transpose<<<cdiv256((long)H * AF),    256, 0, stream>>>(W_i_atom, Wt_ia, H, AF);
  k_transpose<<<cdiv256((long)H * BF),    256, 0, stream>>>(W_i_bond, Wt_ib, H, BF);
  k_transpose<<<cdiv256((long)H * H),     256, 0, stream>>>(W_h,         Wt_h0, H, H);
  k_transpose<<<cdiv256((long)H * H),     256, 0, stream>>>(W_h + H * H, Wt_h1, H, H);
  k_transpose<<<cdiv256((long)H * 3 * H), 256, 0, stream>>>(lr_W,  Wt_lr, H, 3 * H);
  k_transpose<<<cdiv256((long)H * 2 * H), 256, 0, stream>>>(W_o_W, Wt_o,  H, 2 * H);
  k_transpose<<<cdiv256((long)3 * H * H), 256, 0, stream>>>(gru_w_ih,             Wt_ihf, 3 * H, H);
  k_transpose<<<cdiv256((long)3 * H * H), 256, 0, stream>>>(gru_w_ih + 3 * H * H, Wt_ihb, 3 * H, H);
  k_transpose<<<cdiv256((long)3 * H * H), 256, 0, stream>>>(gru_w_hh,             Wt_hhf, 3 * H, H);
  k_transpose<<<cdiv256((long)3 * H * H), 256, 0, stream>>>(gru_w_hh + 3 * H * H, Wt_hhb, 3 * H, H);

  // ---- 2) input projections ----
  launch_gemm(f_atoms, AF, Wt_ia, nullptr, nullptr, iatom, NA, H, AF, 1, stream);
  launch_gemm(f_bonds, BF, Wt_ib, nullptr, nullptr, ibond, NB, H, BF, 1, stream);
  hipMemcpyAsync(matom, iatom, (size_t)NA * H * sizeof(float), hipMemcpyDeviceToDevice, stream);
  hipMemcpyAsync(mbond, ibond, (size_t)NB * H * sizeof(float), hipMemcpyDeviceToDevice, stream);

  // ---- 3) message-passing depth loop ----
  const float* Wt_hd[2] = { Wt_h0, Wt_h1 };
  for (int d = 0; d < 2; ++d) {
    k_aggregate<<<cdiv256((long)NA * H4), 256, 0, stream>>>(mbond, a2b, nullptr, matom, 1);
    k_bond_pre <<<cdiv256((long)NB * H4), 256, 0, stream>>>(matom, mbond, b2a, b2revb, mbtmp);
    launch_gemm(mbtmp, H, Wt_hd[d], ibond, nullptr, mbond, NB, H, H, 1, stream);
  }

  // ---- 4) final aggregate + node projection ----
  k_aggregate<<<cdiv256((long)NA * H4), 256, 0, stream>>>(mbond, a2b, agg, nullptr, 0);
  k_concat3  <<<cdiv256((long)NA * 3 * H4), 256, 0, stream>>>(agg, matom, iatom, C900);
  launch_gemm(C900, 3 * H, Wt_lr, nullptr, nullptr, node, NA, H, 3 * H, 0, stream);

  // ---- 5) GRU prep ----
  k_bias_relu<<<cdiv256((long)NA * H4), 256, 0, stream>>>(node, gru_bias, msg);
  k_h0       <<<cdiv256((long)B * H4),  256, 0, stream>>>(node, h_f, h_b);

  // ---- 6) bidirectional GRU scan: 2 launches per step ----
  const dim3 gru_grid((unsigned)(((B + 63) / 64) * ((3 * H + 63) / 64)), 4);
  for (int step = 0; step < A; ++step) {
    const int tf = step, tb = A - 1 - step;
    k_gemm_gru4<<<gru_grid, 256, 0, stream>>>(
        msg + (size_t)(1 + tf) * H, msg + (size_t)(1 + tb) * H, h_f, h_b,
        A * H, H,
        Wt_ihf, Wt_ihb, Wt_hhf, Wt_hhb,
        gru_b_ih, gru_b_ih + 3 * H, gru_b_hh, gru_b_hh + 3 * H,
        xg_f, xg_b, gh_f, gh_b,
        B, 3 * H, H);
    k_gru_gate<<<cdiv256(2L * B * H), 256, 0, stream>>>(xg_f, gh_f, xg_b, gh_b,
                                                        h_f, h_b, outf, outb, tf, tb);
  }

  // ---- 7) output head ----
  k_gcat<<<cdiv256((long)NA * 2 * H4), 256, 0, stream>>>(msg, outf, outb, gcat);
  launch_gemm(gcat, 2 * H, Wt_o, nullptr, W_o_b, ah, NA, H, 2 * H, 1, stream);
  k_mean<<<cdiv256((long)B * H4), 256, 0, stream>>>(ah, out);
}